// OHZeroBasedClassifier_5093831213833
// MI455X (gfx1250) — compile-verified
//
#include <hip/hip_runtime.h>

// OHZeroBasedClassifier for MI455X (gfx1250).
//
// dist[i][j] = S_i + 1 - 2*x[i][j]  =>  argmin_j dist == argmax_j x[i][j]
// all(x[i] < t) == (max_j x[i][j] < t)
// => pure streaming row-(max,argmax): ~4.16 MB @ 23.3 TB/s ~= 0.18 us floor.
// Data path: TDM tensor_load_to_lds (one DMA per 32x127 f32 tile) + wave32
// shuffle-butterfly argmax reduction. WMMA intentionally unused: no matmul
// survives the algebraic simplification, and a GEMM formulation would double
// memory traffic.

#define THRESHOLD_V     0.24f
#define DIMS            127
#define ROWS_PER_BLOCK  32
#define BLOCK_THREADS   256
#define WAVES_PER_BLOCK (BLOCK_THREADS / 32)
#define ROWS_PER_WAVE   (ROWS_PER_BLOCK / WAVES_PER_BLOCK)

typedef unsigned int       u32;
typedef unsigned long long u64;
typedef u32 __attribute__((ext_vector_type(4))) v4u;
typedef int __attribute__((ext_vector_type(4))) v4i;
typedef int __attribute__((ext_vector_type(8))) v8i;

__global__ __launch_bounds__(BLOCK_THREADS)
void ohz_classifier_kernel(const float* __restrict__ x,
                           float* __restrict__ out,
                           int nrows)
{
    __shared__ float tile[ROWS_PER_BLOCK * DIMS];

    const u32 tid  = threadIdx.x;
    const u32 wave = tid >> 5;
    const u32 lane = tid & 31u;
    const u32 blk  = blockIdx.x;
    const u32 row0 = blk * ROWS_PER_BLOCK;

#if defined(__gfx1250__) && __has_builtin(__builtin_amdgcn_tensor_load_to_lds) && __has_builtin(__builtin_amdgcn_s_wait_tensorcnt)
    // --- Tensor Data Mover: DMA a 32x127 f32 tile (16,256 B) into LDS ---
    if (wave == 0) {  // wave-uniform branch; TDM ignores EXEC, issue once
        const u64 gaddr = (u64)(x + (u64)row0 * DIMS);       // tile start in global
        const u32 ldsa  = (u32)(u64)(&tile[0]);              // low 32b of flat ptr == LDS byte addr

        // D# group 0 (128b): count=1 | lds_addr | global_addr[56:0] | type=2
        v4u g0;
        g0.x = 1u;                                           // count=1 (valid), no gather
        g0.y = ldsa;                                         // lds_addr
        g0.z = (u32)(gaddr & 0xFFFFFFFFull);                 // global_addr[31:0]
        g0.w = ((u32)(gaddr >> 32) & 0x01FFFFFFu)            // global_addr[56:32]
             | 0x80000000u;                                  // type=2 ("image")

        // D# group 1 (256b):
        //  w0: workgroup_mask=0, data_size=2 (4B), no barrier/iterate/pad
        //  tensor_dim0=127, tensor_dim1=nrows, tile_dim0=127, tile_dim1=32,
        //  tile_dim2=0, tensor_dim0_stride=127, tensor_dim1_stride unused (0)
        v8i g1;
        g1[0] = 0x00020000;                                  // data_size = 4 bytes
        g1[1] = (DIMS << 16);                                // tensor_dim0[15:0] @ [31:16]
        g1[2] = ((nrows & 0xFFFF) << 16);                    // tensor_dim0 hi=0 | tensor_dim1 lo
        g1[3] = (DIMS << 16) | ((nrows >> 16) & 0xFFFF);     // tensor_dim1 hi | tile_dim0
        g1[4] = ROWS_PER_BLOCK;                              // tile_dim1 | tile_dim2=0
        g1[5] = DIMS;                                        // tensor_dim0_stride[31:0]
        g1[6] = 0;                                           // stride hi | dim1_stride lo
        g1[7] = 0;

        v4i gz4  = {0, 0, 0, 0};                             // groups 2/3: 2-D tensor, unused
        v8i gz8  = {0, 0, 0, 0, 0, 0, 0, 0};                 // 6-arg form: extra group, unused

        // amdgpu-toolchain (clang-23) 6-arg form:
        //   (v4u g0, v8i g1, v4i g2, v4i g3, v8i g4, i32 cpol)
        __builtin_amdgcn_tensor_load_to_lds(g0, g1, gz4, gz4, gz8, /*cpol=*/0);
        __builtin_amdgcn_s_wait_tensorcnt(0);                // TENSORcnt drained by issuing wave
    }
    __syncthreads();                                          // publish LDS tile to all waves
#else
    // Fallback (host pass / toolchains without the TDM builtin): cooperative copy.
    for (u32 i = tid; i < (u32)(ROWS_PER_BLOCK * DIMS); i += BLOCK_THREADS) {
        tile[i] = x[(u64)row0 * DIMS + i];
    }
    __syncthreads();
#endif

    // --- wave32 (max, argmax) per row; first-index tie-break matches jnp.argmin ---
    for (int k = 0; k < ROWS_PER_WAVE; ++k) {
        const int rloc = (int)wave * ROWS_PER_WAVE + k;
        const float* __restrict__ r = &tile[rloc * DIMS];

        float bestv = -1.0f;   // x >= 0, safe sentinel
        int   besti = 0;

        // lane handles elements {lane, lane+32, lane+64, lane+96<127} in ascending
        // index order; strict '>' keeps the smallest index on ties.
        #pragma unroll
        for (int e = 0; e < 4; ++e) {
            const int idx = (int)lane + 32 * e;
            if (idx < DIMS) {
                const float v = r[idx];
                if (v > bestv) { bestv = v; besti = idx; }
            }
        }

        // 5-step butterfly over the 32 lanes; prefer larger value, then smaller index.
        #pragma unroll
        for (int off = 16; off > 0; off >>= 1) {
            const float ov = __shfl_xor(bestv, off, 32);
            const int   oi = __shfl_xor(besti, off, 32);
            if (ov > bestv || (ov == bestv && oi < besti)) { bestv = ov; besti = oi; }
        }

        if (lane == 0) {
            // all(x < t) <=> max(x) < t  -> class 0; else 1-based argmax.
            out[row0 + rloc] = (bestv < THRESHOLD_V) ? 0.0f : (float)(besti + 1);
        }
    }
}

extern "C" void kernel_launch(void* const* d_in, const int* in_sizes, int n_in,
                              void* d_out, int out_size, void* d_ws, size_t ws_size,
                              hipStream_t stream)
{
    (void)n_in; (void)out_size; (void)d_ws; (void)ws_size;
    const float* x = (const float*)d_in[0];
    // d_in[1] (identity codebook) is unused: L1-to-one-hot reduces to row argmax.
    float* out = (float*)d_out;

    const int nrows   = in_sizes[0] / DIMS;          // 8192
    const int nblocks = nrows / ROWS_PER_BLOCK;      // 256

    ohz_classifier_kernel<<<nblocks, BLOCK_THREADS, 0, stream>>>(x, out, nrows);
}